// NGNN_SAGEConv_28398323761563
// MI455X (gfx1250) — compile-verified
//
#include <hip/hip_runtime.h>

typedef float v2f __attribute__((ext_vector_type(2)));
typedef float v8f __attribute__((ext_vector_type(8)));

#define D    128
#define D4   32     // float4 per feature row
#define AST  132    // padded LDS row stride (floats) for A tiles -> conflict-free A reads
#define AST4 33     // in float4 units

// ---------------------------------------------------------------------------
// Zero the workspace region (agg [N*128] + deg [N]) as float4s.
// ---------------------------------------------------------------------------
__global__ void zero_ws(float4* __restrict__ p, long n4) {
  long i = (long)blockIdx.x * blockDim.x + threadIdx.x;
  if (i < n4) p[i] = make_float4(0.f, 0.f, 0.f, 0.f);
}

// ---------------------------------------------------------------------------
// Edge scatter: one wave32 per edge.  Coalesced float4 row load of x[src],
// 4 global_atomic_add_f32 per lane into agg[dst], lane 0 bumps deg[dst].
// x and agg are L2-resident (51 MB each vs 192 MB L2).
// ---------------------------------------------------------------------------
__global__ void scatter_edges(const float* __restrict__ x,
                              const int* __restrict__ src,
                              const int* __restrict__ dst,
                              float* __restrict__ agg,
                              float* __restrict__ deg,
                              int E) {
  int tid  = blockIdx.x * blockDim.x + threadIdx.x;
  int e    = tid >> 5;
  int lane = tid & 31;
  if (e >= E) return;
  int s = src[e];
  int d = dst[e];
  float4 v = ((const float4*)x)[(long)s * D4 + lane];
  float* a = agg + (long)d * D + lane * 4;
  atomicAdd(a + 0, v.x);
  atomicAdd(a + 1, v.y);
  atomicAdd(a + 2, v.z);
  atomicAdd(a + 3, v.w);
  if (lane == 0) atomicAdd(deg + d, 1.0f);
}

// ---------------------------------------------------------------------------
// Fused GEMM layer:  out = act( A0 @ W0  [+ (A1/deg) @ W1]  + bias )
// Block = 256 threads (8 waves), owns 32 rows x 128 cols.
// Wave w: row-group (w>>2)*16, column group (w&3)*32 -> two 16x16 WMMA tiles.
//
// W is stored K-pair-swizzled in LDS:  W[k][n] -> sW[(k>>1)*256 + n*2 + (k&1)]
// so each lane's B fragment (W[kb][col], W[kb+1][col]) is one ds_load_b64.
// A tiles use a 132-float padded row stride (bank-conflict-free b64 reads).
// K-loop fully unrolled: pure ds_load_b64 -> v_wmma_f32_16x16x4_f32 stream.
// In-place safe: block reads only its own 32 rows (into LDS) before storing.
// ---------------------------------------------------------------------------
template <bool HAS2, bool RELU>
__global__ void __launch_bounds__(256)
gemm_layer(const float* __restrict__ A0, const float* __restrict__ W0,
           const float* __restrict__ A1, const float* __restrict__ W1,
           const float* __restrict__ deg, const float* __restrict__ bias,
           float* __restrict__ out, int nrows) {
  extern __shared__ float lds[];

  float* sW0 = lds;                                    // 16384 floats (swizzled)
  float* sW1 = sW0 + 16384;                            // 16384 floats (if HAS2)
  float* sA0 = HAS2 ? (sW1 + 16384) : (sW0 + 16384);   // 32 * AST floats
  float* sA1 = sA0 + 32 * AST;                         // 32 * AST floats (if HAS2)
  float* sB  = sA0 + (HAS2 ? 2 : 1) * 32 * AST;        // 128 floats

  const int tid  = threadIdx.x;
  const int row0 = blockIdx.x * 32;

  // ---- stage W0 (and W1) swizzled: coalesced global b128 reads ----
  {
    const float4* g = (const float4*)W0;
#pragma unroll
    for (int i = 0; i < 16; ++i) {
      int j = i * 256 + tid;        // float4 index 0..4095
      int k = j >> 5;               // W row
      int n = (j & 31) * 4;         // W col start
      float4 v = g[j];
      float* dp = sW0 + (k >> 1) * 256 + (k & 1);
      dp[(n + 0) * 2] = v.x;
      dp[(n + 1) * 2] = v.y;
      dp[(n + 2) * 2] = v.z;
      dp[(n + 3) * 2] = v.w;
    }
  }
  if constexpr (HAS2) {
    const float4* g = (const float4*)W1;
#pragma unroll
    for (int i = 0; i < 16; ++i) {
      int j = i * 256 + tid;
      int k = j >> 5;
      int n = (j & 31) * 4;
      float4 v = g[j];
      float* dp = sW1 + (k >> 1) * 256 + (k & 1);
      dp[(n + 0) * 2] = v.x;
      dp[(n + 1) * 2] = v.y;
      dp[(n + 2) * 2] = v.z;
      dp[(n + 3) * 2] = v.w;
    }
  }
  // ---- stage A0 tile: 32 rows x 128, padded row stride ----
  {
    const float4* g = (const float4*)A0;
    float4* s = (float4*)sA0;
#pragma unroll
    for (int i = 0; i < 4; ++i) {
      int j  = i * 256 + tid;        // 0..1023
      int r  = j >> 5;               // row in tile
      int c4 = j & 31;               // float4 col
      int gr = row0 + r;
      float4 v = make_float4(0.f, 0.f, 0.f, 0.f);
      if (gr < nrows) v = g[(long)gr * D4 + c4];
      s[r * AST4 + c4] = v;
    }
  }
  // ---- stage A1 tile with 1/max(deg,1) folded in ----
  if constexpr (HAS2) {
    const float4* g = (const float4*)A1;
    float4* s = (float4*)sA1;
#pragma unroll
    for (int i = 0; i < 4; ++i) {
      int j  = i * 256 + tid;
      int r  = j >> 5;
      int c4 = j & 31;
      int gr = row0 + r;
      float4 v = make_float4(0.f, 0.f, 0.f, 0.f);
      if (gr < nrows) {
        v = g[(long)gr * D4 + c4];
        float sc = 1.0f / fmaxf(deg[gr], 1.0f);
        v.x *= sc; v.y *= sc; v.z *= sc; v.w *= sc;
      }
      s[r * AST4 + c4] = v;
    }
  }
  if (tid < 32) ((float4*)sB)[tid] = ((const float4*)bias)[tid];
  __syncthreads();

  const int wave   = tid >> 5;
  const int lane   = tid & 31;
  const int rowgrp = wave >> 2;    // 0/1: rows [row0+16*rowgrp, +16)
  const int wcol   = wave & 3;     // column group of 32
  const int lrow   = lane & 15;    // A row within group / B-N index / D col
  const int khalf  = lane >> 4;    // which K pair this lane holds
  const int c0     = wcol * 32;

  v8f acc0 = {};
  v8f acc1 = {};
  const float* a0base = sA0 + (rowgrp * 16 + lrow) * AST + khalf * 2;
  const float* a1base = sA1 + (rowgrp * 16 + lrow) * AST + khalf * 2;
  const float* w0b0 = sW0 + khalf * 256 + (c0 + lrow) * 2;
  const float* w0b1 = sW0 + khalf * 256 + (c0 + 16 + lrow) * 2;
  const float* w1b0 = sW1 + khalf * 256 + (c0 + lrow) * 2;
  const float* w1b1 = sW1 + khalf * 256 + (c0 + 16 + lrow) * 2;

#pragma unroll
  for (int kk = 0; kk < D; kk += 4) {
    const int kq = kk >> 1;        // K-pair row advance in swizzled W
    v2f a  = *(const v2f*)(a0base + kk);
    v2f b0 = *(const v2f*)(w0b0 + kq * 256);
    v2f b1 = *(const v2f*)(w0b1 + kq * 256);
    acc0 = __builtin_amdgcn_wmma_f32_16x16x4_f32(false, a, false, b0, (short)0, acc0, false, false);
    acc1 = __builtin_amdgcn_wmma_f32_16x16x4_f32(false, a, false, b1, (short)0, acc1, false, false);
    if constexpr (HAS2) {
      v2f a1v = *(const v2f*)(a1base + kk);
      v2f d0  = *(const v2f*)(w1b0 + kq * 256);
      v2f d1  = *(const v2f*)(w1b1 + kq * 256);
      acc0 = __builtin_amdgcn_wmma_f32_16x16x4_f32(false, a1v, false, d0, (short)0, acc0, false, false);
      acc1 = __builtin_amdgcn_wmma_f32_16x16x4_f32(false, a1v, false, d1, (short)0, acc1, false, false);
    }
  }

  // ---- epilogue: D layout -> VGPR g holds row g + 8*khalf, col = lrow ----
  const int r0 = row0 + rowgrp * 16;
#pragma unroll
  for (int g = 0; g < 8; ++g) {
    int m  = g + khalf * 8;
    int gr = r0 + m;
    if (gr < nrows) {
      int col = c0 + lrow;
      float v0 = acc0[g] + sB[col];
      float v1 = acc1[g] + sB[col + 16];
      if constexpr (RELU) { v0 = fmaxf(v0, 0.f); v1 = fmaxf(v1, 0.f); }
      out[(long)gr * D + col]      = v0;
      out[(long)gr * D + col + 16] = v1;
    }
  }
}

// ---------------------------------------------------------------------------
extern "C" void kernel_launch(void* const* d_in, const int* in_sizes, int n_in,
                              void* d_out, int out_size, void* d_ws, size_t ws_size,
                              hipStream_t stream) {
  const float* x   = (const float*)d_in[0];
  const int*   src = (const int*)d_in[1];
  const int*   dst = (const int*)d_in[2];
  const float* Wn  = (const float*)d_in[3];
  const float* Ws  = (const float*)d_in[4];
  const float* bs  = (const float*)d_in[5];
  const float* Wf1 = (const float*)d_in[6];
  const float* bf1 = (const float*)d_in[7];
  const float* Wf2 = (const float*)d_in[8];
  const float* bf2 = (const float*)d_in[9];
  float* out = (float*)d_out;

  const int N = in_sizes[0] / D;
  const int E = in_sizes[1];

  float* agg = (float*)d_ws;               // N*128 floats
  float* deg = agg + (size_t)N * D;        // N floats

  // zero agg + deg
  long n4 = ((long)N * D + (long)N) / 4;
  int  zb = (int)((n4 + 255) / 256);
  zero_ws<<<zb, 256, 0, stream>>>((float4*)d_ws, n4);

  // edge scatter: one wave per edge
  long sthreads = (long)E * 32;
  int  sb = (int)((sthreads + 255) / 256);
  scatter_edges<<<sb, 256, 0, stream>>>(x, src, dst, agg, deg, E);

  const int gb = (N + 31) / 32;
  const size_t sh1 = (size_t)(2 * 16384 + 2 * 32 * AST + 128) * sizeof(float); // ~162 KB
  const size_t sh2 = (size_t)(16384 + 32 * AST + 128) * sizeof(float);         // ~81 KB

  // layer 1: h = relu(x@W_self + b_self + (agg/deg)@W_neigh)   (in-place -> agg)
  gemm_layer<true, true><<<gb, 256, sh1, stream>>>(x, Ws, agg, Wn, deg, bs, agg, N);
  // layer 2: h = relu(h@W_fc1 + b_fc1)                         (in-place -> agg)
  gemm_layer<false, true><<<gb, 256, sh2, stream>>>(agg, Wf1, nullptr, nullptr, nullptr, bf1, agg, N);
  // layer 3: out = h@W_fc2 + b_fc2
  gemm_layer<false, false><<<gb, 256, sh2, stream>>>(agg, Wf2, nullptr, nullptr, nullptr, bf2, out, N);
}